// LocalMessagePassingNetwork_2559800508842
// MI455X (gfx1250) — compile-verified
//
#include <hip/hip_runtime.h>

typedef __attribute__((ext_vector_type(2))) float v2f;
typedef __attribute__((ext_vector_type(8))) float v8f;

#define DIM        128
#define NEDGES     50000
#define NEG_SLOPE  0.01f

// ---------------------------------------------------------------------------
// K1: Y = X @ W via V_WMMA_F32_16X16X4_F32.  One wave per 16x16 output tile.
// A (16x4 f32): lanes 0-15 = M, vgpr0/1 = K0/K1; lanes 16-31 = K2/K3.
// B (4x16 f32): lanes 0-15 = N, vgpr0/1 = K0/K1; lanes 16-31 = K2/K3.
// C/D (16x16 f32): vgpr r -> M=r (lanes 0-15) / M=8+r (lanes 16-31), N=lane%16.
// ---------------------------------------------------------------------------
__global__ void __launch_bounds__(256)
gemm_wmma_f32(const float* __restrict__ X, const float* __restrict__ W,
              float* __restrict__ Y, int mtiles)
{
    const int lane  = threadIdx.x & 31;
    const int wv    = threadIdx.x >> 5;
    const long tile = (long)blockIdx.x * 8 + wv;   // 8 waves per block
    const long mt   = tile >> 3;                   // 8 column tiles (DIM/16)
    const int  nt   = (int)(tile & 7);
    if (mt >= mtiles) return;                      // wave-uniform

    const int half = lane >> 4;
    const int lr   = lane & 15;
    const size_t row  = (size_t)mt * 16 + lr;
    const int    col0 = nt * 16;

    const float* __restrict__ ap = X + row * DIM + 2 * half;
    const float* __restrict__ bp = W + (size_t)(2 * half) * DIM + col0 + lr;

    v8f c = {};
#pragma unroll
    for (int k = 0; k < DIM; k += 4) {
        v2f a, b;
        a.x = ap[k];
        a.y = ap[k + 1];
        b.x = bp[(size_t)k * DIM];
        b.y = bp[(size_t)k * DIM + DIM];
        c = __builtin_amdgcn_wmma_f32_16x16x4_f32(false, a, false, b,
                                                  (short)0, c, false, false);
    }

    float* __restrict__ yp = Y + ((size_t)mt * 16 + half * 8) * DIM + col0 + lr;
#pragma unroll
    for (int r = 0; r < 8; ++r)
        yp[(size_t)r * DIM] = c[r];
}

// ---------------------------------------------------------------------------
__global__ void zero_kernel(float* __restrict__ p, long n)
{
    long i = (long)blockIdx.x * blockDim.x + threadIdx.x;
    if (i < n) p[i] = 0.0f;
}

// K3: node/edge degrees via hardware f32 atomics (resolve in L2 atomic units)
__global__ void degrees_kernel(const int* __restrict__ node,
                               const int* __restrict__ edge,
                               float* __restrict__ Ddeg,
                               float* __restrict__ Bdeg, int nnz)
{
    int i = blockIdx.x * blockDim.x + threadIdx.x;
    if (i >= nnz) return;
    unsafeAtomicAdd(&Ddeg[node[i]], 1.0f);
    unsafeAtomicAdd(&Bdeg[edge[i]], 1.0f);
}

// K4/K7: one wave per incidence entry; each lane moves a float4 (512B/wave
// gather via global_load_b128, scatter via global_atomic_add_f32).
__global__ void __launch_bounds__(256)
scatter_kernel(const float* __restrict__ src, float* __restrict__ dst,
               const int* __restrict__ gidx, const int* __restrict__ sidx,
               int nnz)
{
    const int wid  = (int)(((long)blockIdx.x * blockDim.x + threadIdx.x) >> 5);
    const int lane = threadIdx.x & 31;
    if (wid >= nnz) return;                        // wave-uniform

    const int g = gidx[wid];
    const int s = sidx[wid];

    const float4 v = ((const float4*)(src + (size_t)g * DIM))[lane];
    float* dp = dst + (size_t)s * DIM + lane * 4;
    unsafeAtomicAdd(dp + 0, v.x);
    unsafeAtomicAdd(dp + 1, v.y);
    unsafeAtomicAdd(dp + 2, v.z);
    unsafeAtomicAdd(dp + 3, v.w);
}

// K5: m[e, :] *= 1/B[e]  (0 if B==0)
__global__ void scale_rows_kernel(float* __restrict__ m,
                                  const float* __restrict__ Bdeg, long n)
{
    long i = (long)blockIdx.x * blockDim.x + threadIdx.x;
    if (i >= n) return;
    float bb  = Bdeg[i >> 7];
    float inv = bb > 0.0f ? 1.0f / bb : 0.0f;
    m[i] *= inv;
}

// K8: out = LeakyReLU(out * Dinv + bias)
__global__ void finalize_kernel(float* __restrict__ out,
                                const float* __restrict__ Ddeg,
                                const float* __restrict__ bias, long n)
{
    long i = (long)blockIdx.x * blockDim.x + threadIdx.x;
    if (i >= n) return;
    float dd  = Ddeg[i >> 7];
    float inv = dd > 0.0f ? 1.0f / dd : 0.0f;
    float v   = out[i] * inv + bias[i & (DIM - 1)];
    out[i] = v >= 0.0f ? v : NEG_SLOPE * v;
}

// ---------------------------------------------------------------------------
extern "C" void kernel_launch(void* const* d_in, const int* in_sizes, int n_in,
                              void* d_out, int out_size, void* d_ws, size_t ws_size,
                              hipStream_t stream)
{
    const float* X    = (const float*)d_in[0];
    const int*   A    = (const int*)d_in[1];
    const float* W    = (const float*)d_in[2];
    const float* bias = (const float*)d_in[3];
    float*       out  = (float*)d_out;

    const int N   = in_sizes[0] / DIM;   // 200000 nodes
    const int NNZ = in_sizes[1] / 2;     // 2000000 incidence entries
    const int* node = A;                 // A[0, :]
    const int* edge = A + NNZ;           // A[1, :]

    // workspace layout: m [NEDGES*DIM] | D [N] | B [NEDGES]   (~27 MB)
    float* m    = (float*)d_ws;
    float* Ddeg = m + (size_t)NEDGES * DIM;
    float* Bdeg = Ddeg + N;
    const long ws_f  = (long)NEDGES * DIM + N + NEDGES;
    const long n_out = (long)N * DIM;
    const long n_m   = (long)NEDGES * DIM;

    // K1: x = X @ W   (stored in d_out, consumed by pass 1)
    const int mtiles = N / 16;
    gemm_wmma_f32<<<mtiles, 256, 0, stream>>>(X, W, out, mtiles);

    // K2: zero workspace (m, D, B are contiguous)
    zero_kernel<<<(int)((ws_f + 255) / 256), 256, 0, stream>>>(m, ws_f);

    // K3: degrees
    degrees_kernel<<<(NNZ + 255) / 256, 256, 0, stream>>>(node, edge, Ddeg, Bdeg, NNZ);

    // K4: pass 1, nodes -> hyperedges: m[edge] += x[node]
    scatter_kernel<<<(NNZ + 7) / 8, 256, 0, stream>>>(out, m, node, edge, NNZ);

    // K5: m *= Binv
    scale_rows_kernel<<<(int)((n_m + 255) / 256), 256, 0, stream>>>(m, Bdeg, n_m);

    // K6: reuse d_out as pass-2 accumulator
    zero_kernel<<<(int)((n_out + 255) / 256), 256, 0, stream>>>(out, n_out);

    // K7: pass 2, hyperedges -> nodes: out[node] += m[edge]
    scatter_kernel<<<(NNZ + 7) / 8, 256, 0, stream>>>(m, out, edge, node, NNZ);

    // K8: out = LeakyReLU(out * Dinv + b)
    finalize_kernel<<<(int)((n_out + 255) / 256), 256, 0, stream>>>(out, Ddeg, bias, n_out);
}